// MultiBoxEL_89352499626003
// MI455X (gfx1250) — compile-verified
//
#include <hip/hip_runtime.h>
#include <math.h>

// ---------------------------------------------------------------------------
// MultiBoxEL nf1_loss for MI455X (gfx1250, wave32).
// Gather-bound problem: 20MB embedding table lives in the 192MB L2; one pair
// per thread, unrolled 25-dim product loop. Deterministic block reduction via
// chained V_WMMA_F32_16X16X4_F32 (B = ones => row-sum), exact f32 arithmetic.
// ---------------------------------------------------------------------------

typedef float v2f __attribute__((ext_vector_type(2)));
typedef float v8f __attribute__((ext_vector_type(8)));

#define EMB_BOUND 10000.0f
#define DIMH 25
#define ROW 50
#define NBLOCKS 1024
#define NTHREADS 256

// Reduce 256 per-thread values (one per thread of the block) to a single f32
// sum using V_WMMA_F32_16X16X4_F32. All waves run the identical WMMA sequence
// on the shared buffer (uniform control flow => EXEC all-ones, as the ISA
// requires); the redundancy across 8 waves is negligible.
//
// A-matrix (16x4 f32, wave32 layout per ISA 7.12.2):
//   lanes 0-15 : VGPR0 = A[m=lane][k=0], VGPR1 = A[m=lane][k=1]
//   lanes 16-31: VGPR0 = A[m=lane-16][k=2], VGPR1 = A[m=lane-16][k=3]
// With B = ones(4x16):  D[m][n] = sum_k A[m][k]  (all columns identical).
// Chunk ch covers buf[ch*64 .. ch*64+63] with A[m][k] = buf[ch*64 + m*4 + k];
// accumulating through C over 4 chunks sums all 256 values into row sums.
// Per-lane fold of the 8 D VGPRs gives rows 0-7 (lanes 0-15) / rows 8-15
// (lanes 16-31) at one column; shfl_xor(16) completes the total in every lane.
__device__ __forceinline__ float block_reduce_256_wmma(float v) {
  __shared__ __align__(16) float buf[NTHREADS];
  buf[threadIdx.x] = v;
  __syncthreads();

  const int l = threadIdx.x & 31;
  v8f c = {0.f, 0.f, 0.f, 0.f, 0.f, 0.f, 0.f, 0.f};
  v2f ones;
  ones[0] = 1.0f;
  ones[1] = 1.0f;
#pragma unroll
  for (int ch = 0; ch < 4; ++ch) {
    const int idx = ch * 64 + (l & 15) * 4 + ((l & 16) ? 2 : 0);
    v2f a;
    a[0] = buf[idx];
    a[1] = buf[idx + 1];
    // D = A(16x4) * ones(4x16) + C   ->  v_wmma_f32_16x16x4_f32
    c = __builtin_amdgcn_wmma_f32_16x16x4_f32(
        /*neg_a=*/false, a, /*neg_b=*/false, ones,
        /*c_mod=*/(short)0, c, /*reuse_a=*/false, /*reuse_b=*/false);
  }
  float s = c[0] + c[1] + c[2] + c[3] + c[4] + c[5] + c[6] + c[7];
  s += __shfl_xor(s, 16, 32);  // rows0-7 + rows8-15 -> full 256-sum, all lanes
  return s;
}

__global__ void __launch_bounds__(NTHREADS)
nf1_partial_kernel(const float* __restrict__ ce, const int* __restrict__ nf,
                   float* __restrict__ partial, int npairs) {
  float acc = 0.f;
  const int stride = gridDim.x * blockDim.x;
  for (int i = blockIdx.x * blockDim.x + threadIdx.x; i < npairs; i += stride) {
    const int ci = nf[2 * i];
    const int di = nf[2 * i + 1];
    const float* __restrict__ cr = ce + ci * ROW;
    const float* __restrict__ dr = ce + di * ROW;
    float inter = 1.f;
    float carea = 1.f;
#pragma unroll
    for (int j = 0; j < DIMH; ++j) {
      const float cc = cr[j];
      const float co = fabsf(cr[j + DIMH]);
      const float dc = dr[j];
      const float df = fabsf(dr[j + DIMH]);
      const float lo = fmaxf(cc - co, dc - df);
      const float hi = fminf(cc + co, dc + df);
      inter *= fmaxf(hi - lo, 0.f);
      carea *= 2.f * co;
    }
    float loss;
    if (carea == 0.f) {
      loss = 0.f;
    } else if (__builtin_isinf(carea)) {
      loss = 1.f - inter / (2.f * EMB_BOUND);
    } else {
      loss = 1.f - inter / carea;
    }
    const float r = fmaxf(loss, 0.f);
    acc = fmaf(r, r, acc);
  }
  const float tot = block_reduce_256_wmma(acc);
  if (threadIdx.x == 0) partial[blockIdx.x] = tot;
}

__global__ void __launch_bounds__(NTHREADS)
nf1_final_kernel(const float* __restrict__ partial, float* __restrict__ out,
                 int nparts) {
  float v = 0.f;
  for (int i = threadIdx.x; i < nparts; i += NTHREADS) v += partial[i];
  const float tot = block_reduce_256_wmma(v);
  if (threadIdx.x == 0) out[0] = sqrtf(tot);
}

extern "C" void kernel_launch(void* const* d_in, const int* in_sizes, int n_in,
                              void* d_out, int out_size, void* d_ws,
                              size_t ws_size, hipStream_t stream) {
  const float* ce = (const float*)d_in[0];  // class_embeds: (100000, 50) f32
  const int* nf = (const int*)d_in[1];      // nf1_data: (2000000, 2) i32
  const int npairs = in_sizes[1] / 2;
  float* partials = (float*)d_ws;           // NBLOCKS f32 (4 KB) scratch

  nf1_partial_kernel<<<NBLOCKS, NTHREADS, 0, stream>>>(ce, nf, partials,
                                                       npairs);
  nf1_final_kernel<<<1, NTHREADS, 0, stream>>>(partials, (float*)d_out,
                                               NBLOCKS);
}